// Splatter_40991167873409
// MI455X (gfx1250) — compile-verified
//
#include <hip/hip_runtime.h>
#include <hip/hip_bf16.h>
#include <stdint.h>

#define N_G    10000
#define W_IMG  256
#define H_IMG  256
#define FX     200.0f
#define FY     200.0f
#define TILE   16
#define NTX    (W_IMG / TILE)
#define NTY    (H_IMG / TILE)
#define NT     (NTX * NTY)
#define K_SEL  256
#define NEAR_P 0.3f
#define MAGIC  1.2f
#define LPAD   (0.3f / (FX * FX))
#define HALF_W (W_IMG * MAGIC * 0.5f / FX)
#define HALF_H (H_IMG * MAGIC * 0.5f / FY)
#define NP     10240   /* padded SoA stride in workspace (floats) */
#define WS_PK  (11 * NP) /* offset (floats) of packed float4 {mx,my,rad,z} */

typedef __attribute__((ext_vector_type(2))) float v2f;
typedef __attribute__((ext_vector_type(8))) float v8f;

__device__ __forceinline__ float sigmoidf_(float x) { return 1.0f / (1.0f + expf(-x)); }

// ---------------------------------------------------------------------------
// Kernel 1: per-gaussian preprocessing -> SoA workspace
// ws layout (stride NP floats): 0:mx 1:my 2:z 3:a 4:b 5:c 6:opa 7:r 8:g 9:b 10:rad
// plus packed float4 {mx,my,rad,z} at WS_PK for the hot distance scan.
// ---------------------------------------------------------------------------
__global__ __launch_bounds__(256) void preprocess_kernel(
    const float* __restrict__ pos, const float* __restrict__ rgb,
    const float* __restrict__ opac, const float* __restrict__ quat,
    const float* __restrict__ scal, const float* __restrict__ w2cr,
    const float* __restrict__ w2ct, float* __restrict__ ws) {
  const int n = blockIdx.x * blockDim.x + threadIdx.x;
  if (n >= N_G) return;

  float qw = quat[4 * n + 0], qx = quat[4 * n + 1];
  float qy = quat[4 * n + 2], qz = quat[4 * n + 3];
  const float qinv = 1.0f / sqrtf(qw * qw + qx * qx + qy * qy + qz * qz);
  qw *= qinv; qx *= qinv; qy *= qinv; qz *= qinv;

  float R[3][3];
  R[0][0] = 1.f - 2.f * (qy * qy + qz * qz);
  R[0][1] = 2.f * (qx * qy - qw * qz);
  R[0][2] = 2.f * (qx * qz + qw * qy);
  R[1][0] = 2.f * (qx * qy + qw * qz);
  R[1][1] = 1.f - 2.f * (qx * qx + qz * qz);
  R[1][2] = 2.f * (qy * qz - qw * qx);
  R[2][0] = 2.f * (qx * qz - qw * qy);
  R[2][1] = 2.f * (qy * qz + qw * qx);
  R[2][2] = 1.f - 2.f * (qx * qx + qy * qy);

  float s2[3];
#pragma unroll
  for (int k = 0; k < 3; ++k) { float s = expf(scal[3 * n + k]); s2[k] = s * s; }

  float C3[3][3];
#pragma unroll
  for (int i = 0; i < 3; ++i)
#pragma unroll
    for (int j = 0; j < 3; ++j)
      C3[i][j] = R[i][0] * s2[0] * R[j][0] + R[i][1] * s2[1] * R[j][1] + R[i][2] * s2[2] * R[j][2];

  float Wr[3][3], Wt[3];
#pragma unroll
  for (int i = 0; i < 3; ++i) {
    Wt[i] = w2ct[i];
#pragma unroll
    for (int j = 0; j < 3; ++j) Wr[i][j] = w2cr[3 * i + j];
  }

  const float p0 = pos[3 * n + 0], p1 = pos[3 * n + 1], p2 = pos[3 * n + 2];
  float pc[3];
#pragma unroll
  for (int i = 0; i < 3; ++i) pc[i] = Wr[i][0] * p0 + Wr[i][1] * p1 + Wr[i][2] * p2 + Wt[i];
  const float x = pc[0], y = pc[1], z = pc[2];
  const float zc = fmaxf(z, 1e-6f);
  const float mx = x / zc, my = y / zc;
  const float iz = 1.0f / zc;

  // cov_cam = Wr * C3 * Wr^T
  float M1[3][3];
#pragma unroll
  for (int i = 0; i < 3; ++i)
#pragma unroll
    for (int j = 0; j < 3; ++j)
      M1[i][j] = Wr[i][0] * C3[0][j] + Wr[i][1] * C3[1][j] + Wr[i][2] * C3[2][j];
  float CC[3][3];
#pragma unroll
  for (int i = 0; i < 3; ++i)
#pragma unroll
    for (int j = 0; j < 3; ++j)
      CC[i][j] = M1[i][0] * Wr[j][0] + M1[i][1] * Wr[j][1] + M1[i][2] * Wr[j][2];

  // J = [[iz,0,-x iz^2],[0,iz,-y iz^2]];  cov2d = J CC J^T
  const float J00 = iz, J02 = -x * iz * iz, J11 = iz, J12 = -y * iz * iz;
  const float t00 = J00 * CC[0][0] + J02 * CC[2][0];
  const float t01 = J00 * CC[0][1] + J02 * CC[2][1];
  const float t02 = J00 * CC[0][2] + J02 * CC[2][2];
  const float t11 = J11 * CC[1][1] + J12 * CC[2][1];
  const float t12 = J11 * CC[1][2] + J12 * CC[2][2];
  const float a = t00 * J00 + t02 * J02 + LPAD;
  const float b = t01 * J11 + t02 * J12;
  const float c = t11 * J11 + t12 * J12 + LPAD;

  const float mid = 0.5f * (a + c);
  const float dd = 0.5f * (a - c);
  const float sq = sqrtf(fmaxf(dd * dd + b * b, 0.0f));
  const float sigma_max = sqrtf(fmaxf(mid + sq, 1e-12f));
  const bool in_fr = (z > NEAR_P) && (fabsf(mx) < HALF_W) && (fabsf(my) < HALF_H);
  const float tile_half_diag =
      0.5f * sqrtf((TILE / FX) * (TILE / FX) + (TILE / FY) * (TILE / FY));
  const float rad = in_fr ? (3.0f * sigma_max + tile_half_diag) : -1.0f;

  ws[0 * NP + n] = mx;
  ws[1 * NP + n] = my;
  ws[2 * NP + n] = z;
  ws[3 * NP + n] = a;
  ws[4 * NP + n] = b;
  ws[5 * NP + n] = c;
  ws[6 * NP + n] = sigmoidf_(opac[n]);
  ws[7 * NP + n] = sigmoidf_(rgb[3 * n + 0]);
  ws[8 * NP + n] = sigmoidf_(rgb[3 * n + 1]);
  ws[9 * NP + n] = sigmoidf_(rgb[3 * n + 2]);
  ws[10 * NP + n] = rad;
  float4 pk;
  pk.x = mx; pk.y = my; pk.z = rad; pk.w = z;
  ((float4*)(ws + WS_PK))[n] = pk;
}

// ---------------------------------------------------------------------------
// Kernel 2: one block per tile. Select top-K nearest valid gaussians (exact
// radix select + stable compaction), depth-sort them (bitonic), then render
// 256 pixels with the weight x rgb GEMM executed on v_wmma_f32_16x16x4_f32.
// ---------------------------------------------------------------------------
__global__ __launch_bounds__(256) void render_kernel(const float* __restrict__ ws,
                                                     float* __restrict__ out) {
  const int tid = threadIdx.x;
  const int t = blockIdx.x;
  const int tx = t % NTX, ty = t / NTX;
  const float cx = (tx * TILE + TILE * 0.5f - W_IMG * 0.5f) / FX;
  const float cy = (ty * TILE + TILE * 0.5f - H_IMG * 0.5f) / FY;

  __shared__ uint32_t s_keys[N_G];        // dist keys; later aliased as weight buffer
  __shared__ uint32_t s_hist[256];
  __shared__ uint32_t s_scanL[256];
  __shared__ uint32_t s_scanE[256];
  __shared__ int s_sel[K_SEL];            // selection, later reused as sort-permutation
  __shared__ float s_gmx[K_SEL], s_gmy[K_SEL], s_ga[K_SEL], s_gb[K_SEL], s_gc[K_SEL];
  __shared__ float s_gop[K_SEL], s_gcr[K_SEL], s_gcg[K_SEL], s_gcb[K_SEL], s_gvl[K_SEL];
  __shared__ unsigned long long s_sort[K_SEL];
  __shared__ uint32_t s_bc[2];

  const float* wmx = ws + 0 * NP;
  const float* wmy = ws + 1 * NP;
  const float* wz  = ws + 2 * NP;
  const float* wa  = ws + 3 * NP;
  const float* wb  = ws + 4 * NP;
  const float* wc  = ws + 5 * NP;
  const float* wop = ws + 6 * NP;
  const float* wcr = ws + 7 * NP;
  const float* wcg = ws + 8 * NP;
  const float* wcb = ws + 9 * NP;
  const float4* wpk = (const float4*)(ws + WS_PK);

  // ---- phase 1: distance keys (monotone uint keys; invalid -> 0xFFFFFFFF)
  for (int i = tid; i < N_G; i += 256) {
    const float4 pk = wpk[i];  // {mx, my, rad, z} in one global_load_b128
    const float dx = cx - pk.x;
    const float dy = cy - pk.y;
    const float dist = sqrtf(dx * dx + dy * dy);
    s_keys[i] = (dist <= pk.z) ? __float_as_uint(dist) : 0xFFFFFFFFu;
  }
  __syncthreads();

  // ---- phase 2: exact K-th smallest key via 4x8-bit radix select
  uint32_t prefix = 0;
  int want = K_SEL;
  for (int pass = 0; pass < 4; ++pass) {
    const int shift = 24 - 8 * pass;
    s_hist[tid] = 0;
    __syncthreads();
    for (int i = tid; i < N_G; i += 256) {
      const uint32_t key = s_keys[i];
      const bool match = (pass == 0) || ((key >> (shift + 8)) == prefix);
      if (match) atomicAdd(&s_hist[(key >> shift) & 0xFFu], 1u);
    }
    __syncthreads();
    // parallel inclusive scan of the 256 histogram bins
    const uint32_t mybin = s_hist[tid];
    s_scanL[tid] = mybin;
    __syncthreads();
    for (int off = 1; off < 256; off <<= 1) {
      uint32_t vv = 0;
      if (tid >= off) vv = s_scanL[tid - off];
      __syncthreads();
      s_scanL[tid] += vv;
      __syncthreads();
    }
    const uint32_t incl = s_scanL[tid];
    const uint32_t excl = incl - mybin;
    if (excl < (uint32_t)want && (uint32_t)want <= incl) {
      s_bc[0] = (uint32_t)tid;  // digit
      s_bc[1] = excl;           // count below digit
    }
    __syncthreads();
    want -= (int)s_bc[1];
    prefix = (prefix << 8) | s_bc[0];
    __syncthreads();
  }
  const uint32_t pivot = prefix;
  const int takeEq = want;

  // ---- phase 3: index-stable compaction of the K selected gaussians
  const int chunk = (N_G + 255) / 256;
  const int beg = tid * chunk;
  const int end = min(beg + chunk, N_G);
  uint32_t cl = 0, ce = 0;
  for (int i = beg; i < end; ++i) {
    const uint32_t key = s_keys[i];
    cl += (key < pivot);
    ce += (key == pivot);
  }
  s_scanL[tid] = cl;
  s_scanE[tid] = ce;
  __syncthreads();
  for (int off = 1; off < 256; off <<= 1) {
    uint32_t vl = 0, ve = 0;
    if (tid >= off) { vl = s_scanL[tid - off]; ve = s_scanE[tid - off]; }
    __syncthreads();
    s_scanL[tid] += vl;
    s_scanE[tid] += ve;
    __syncthreads();
  }
  const uint32_t totalL = s_scanL[255];
  uint32_t posL = s_scanL[tid] - cl;
  uint32_t posE = s_scanE[tid] - ce;
  s_sel[tid] = 0;
  __syncthreads();
  for (int i = beg; i < end; ++i) {
    const uint32_t key = s_keys[i];
    if (key < pivot) {
      s_sel[posL++] = i;
    } else if (key == pivot) {
      if ((int)posE < takeEq) s_sel[totalL + posE] = i;
      posE++;
    }
  }
  __syncthreads();

  // ---- phase 4: gather + 64-bit depth keys (invalid -> +inf)
  {
    const int g = s_sel[tid];
    const uint32_t key = s_keys[g];
    const bool vld = (key != 0xFFFFFFFFu);
    s_gmx[tid] = wmx[g];
    s_gmy[tid] = wmy[g];
    s_ga[tid] = wa[g];
    s_gb[tid] = wb[g];
    s_gc[tid] = wc[g];
    s_gop[tid] = wop[g];
    s_gcr[tid] = wcr[g];
    s_gcg[tid] = wcg[g];
    s_gcb[tid] = wcb[g];
    s_gvl[tid] = vld ? 1.0f : 0.0f;
    const uint32_t zb = vld ? __float_as_uint(wz[g]) : 0x7F800000u;
    s_sort[tid] = ((unsigned long long)zb << 32) | (unsigned long long)(uint32_t)tid;
  }
  __syncthreads();

  // ---- phase 5: bitonic sort (256 elems, ascending by (z, slot))
  for (unsigned k = 2; k <= (unsigned)K_SEL; k <<= 1) {
    for (unsigned j = k >> 1; j > 0; j >>= 1) {
      const unsigned ixj = (unsigned)tid ^ j;
      if (ixj > (unsigned)tid) {
        const unsigned long long va = s_sort[tid];
        const unsigned long long vb = s_sort[ixj];
        const bool asc = (((unsigned)tid & k) == 0);
        if (asc ? (va > vb) : (va < vb)) {
          s_sort[tid] = vb;
          s_sort[ixj] = va;
        }
      }
      __syncthreads();
    }
  }
  s_sel[tid] = (int)(uint32_t)(s_sort[tid] & 0xFFFFFFFFull);  // reuse as perm
  __syncthreads();
  const int* s_perm = s_sel;

  // ---- phase 6: render. Per-thread pixel weights -> LDS -> WMMA blend GEMM.
  // wbuf rows read by wave w (rows 32w..32w+31) are written only by wave w's
  // own threads, and CDNA5 keeps a wave's LDS ops in order (DScnt), so no
  // block barrier is needed inside this loop.
  const int xx = tid & (TILE - 1);
  const int yy = tid >> 4;
  const float u = (tx * TILE + xx + 0.5f - W_IMG * 0.5f) / FX;
  const float v = (ty * TILE + yy + 0.5f - H_IMG * 0.5f) / FY;
  float T = 1.0f;
  float* wbuf = (float*)s_keys;  // [256][17], aliases the (dead) key buffer
  const int lane = tid & 31;
  const int wave = tid >> 5;
  const int r = lane & 15;   // row within M-block / column N for B
  const int hi = lane >> 4;  // which K half (A/B layout) and M offset (C/D layout)
  const int m0 = 2 * wave, m1 = 2 * wave + 1;

  v8f acc0 = {0.f, 0.f, 0.f, 0.f, 0.f, 0.f, 0.f, 0.f};
  v8f acc1 = {0.f, 0.f, 0.f, 0.f, 0.f, 0.f, 0.f, 0.f};

  for (int kb = 0; kb < K_SEL; kb += 16) {
#pragma unroll
    for (int kk = 0; kk < 16; ++kk) {
      const int slot = s_perm[kb + kk];
      const float dx = u - s_gmx[slot];
      const float dy = v - s_gmy[slot];
      const float a = s_ga[slot], b = s_gb[slot], c = s_gc[slot];
      const float det = fmaxf(a * c - b * b, 1e-12f);
      const float maha = (c * dx * dx - 2.0f * b * dx * dy + a * dy * dy) / det;
      const float alpha =
          fminf(fmaxf(s_gop[slot] * expf(-0.5f * maha), 0.0f), 0.99f) * s_gvl[slot];
      wbuf[tid * 17 + kk] = alpha * T;
      T *= (1.0f - alpha);
    }
#pragma unroll
    for (int ks = 0; ks < 16; ks += 4) {
      // B (4x16, K x N): VGPR0 = rows {k0,k0+2}, VGPR1 = rows {k0+1,k0+3}
      const int gi = kb + ks + (hi ? 2 : 0);
      const int sB0 = s_perm[gi];
      const int sB1 = s_perm[gi + 1];
      float b0 = 0.0f, b1 = 0.0f;
      if (r == 0) { b0 = s_gcr[sB0]; b1 = s_gcr[sB1]; }
      else if (r == 1) { b0 = s_gcg[sB0]; b1 = s_gcg[sB1]; }
      else if (r == 2) { b0 = s_gcb[sB0]; b1 = s_gcb[sB1]; }
      v2f Bv;
      Bv.x = b0;
      Bv.y = b1;
      // A (16x4, M x K): lanes 0-15 K={k0,k0+1}, lanes 16-31 K={k0+2,k0+3}
      const int kA = ks + (hi ? 2 : 0);
      v2f A0, A1;
      A0.x = wbuf[(m0 * 16 + r) * 17 + kA];
      A0.y = wbuf[(m0 * 16 + r) * 17 + kA + 1];
      A1.x = wbuf[(m1 * 16 + r) * 17 + kA];
      A1.y = wbuf[(m1 * 16 + r) * 17 + kA + 1];
      acc0 = __builtin_amdgcn_wmma_f32_16x16x4_f32(false, A0, false, Bv, (short)0, acc0,
                                                   false, false);
      acc1 = __builtin_amdgcn_wmma_f32_16x16x4_f32(false, A1, false, Bv, (short)0, acc1,
                                                   false, false);
    }
  }

  // ---- phase 7: writeout. C/D layout: VGPR j -> M=j (lanes0-15) / M=j+8 (16-31)
  if (r < 3) {
#pragma unroll
    for (int j = 0; j < 8; ++j) {
      const int M = j + hi * 8;
      int p = m0 * 16 + M;
      int px = p & 15, py = p >> 4;
      out[((ty * TILE + py) * W_IMG + (tx * TILE + px)) * 3 + r] =
          fminf(fmaxf(acc0[j], 0.0f), 1.0f);
      p = m1 * 16 + M;
      px = p & 15;
      py = p >> 4;
      out[((ty * TILE + py) * W_IMG + (tx * TILE + px)) * 3 + r] =
          fminf(fmaxf(acc1[j], 0.0f), 1.0f);
    }
  }
}

// ---------------------------------------------------------------------------
extern "C" void kernel_launch(void* const* d_in, const int* in_sizes, int n_in,
                              void* d_out, int out_size, void* d_ws, size_t ws_size,
                              hipStream_t stream) {
  (void)in_sizes; (void)n_in; (void)out_size; (void)ws_size;
  const float* pos  = (const float*)d_in[0];
  const float* rgb  = (const float*)d_in[1];
  const float* opac = (const float*)d_in[2];
  const float* quat = (const float*)d_in[3];
  const float* scal = (const float*)d_in[4];
  const float* w2cr = (const float*)d_in[5];
  const float* w2ct = (const float*)d_in[6];
  float* out = (float*)d_out;
  float* ws = (float*)d_ws;

  preprocess_kernel<<<(N_G + 255) / 256, 256, 0, stream>>>(pos, rgb, opac, quat, scal,
                                                           w2cr, w2ct, ws);
  render_kernel<<<NT, 256, 0, stream>>>(ws, out);
}